// LossUResidu_88167088652922
// MI455X (gfx1250) — compile-verified
//
#include <hip/hip_runtime.h>
#include <cstdint>
#include <cstddef>

// Problem constants (match reference)
#define N      128
#define NN     (N * N)            // 16384
#define NNN    (N * N * N)        // 2097152
#define BATCH  4
#define TY     16                 // y-tile per block
#define YT     (N / TY)           // 8 y-tiles
#define BLOCKS_PER_SAMPLE (N * YT)       // 1024
#define NBLOCKS (BATCH * BLOCKS_PER_SAMPLE) // 4096
#define THREADS 256               // 8 wave32 waves

// ---- CDNA5 async global->LDS copy (ASYNCcnt path), with compile-safe fallback ----
typedef int v4i __attribute__((vector_size(4 * sizeof(int))));
typedef __attribute__((address_space(1))) v4i* gv4i_ptr; // global int4*
typedef __attribute__((address_space(3))) v4i* lv4i_ptr; // LDS int4*

__device__ __forceinline__ void async_copy16(const float* g, float* l) {
#if __has_builtin(__builtin_amdgcn_global_load_async_to_lds_b128)
    // emits GLOBAL_LOAD_ASYNC_TO_LDS_B128 (tracked with ASYNCcnt)
    __builtin_amdgcn_global_load_async_to_lds_b128(
        (gv4i_ptr)(void*)g, (lv4i_ptr)(void*)l, 0, 0);
#else
    *(float4*)l = *(const float4*)g; // fallback: load + ds_store (DScnt)
#endif
}

__device__ __forceinline__ void wait_async_zero() {
#if __has_builtin(__builtin_amdgcn_s_wait_asynccnt)
    __builtin_amdgcn_s_wait_asynccnt(0);
#else
    asm volatile("s_wait_asynccnt 0" ::: "memory");
#endif
}

// ---- Main pass: per-block partial sum of |lap3d(u)|*(img==0) and partial max of diag+fde ----
__global__ __launch_bounds__(THREADS) void lossu_stencil_kernel(
        const float* __restrict__ u, const float* __restrict__ img,
        float* __restrict__ psum, float* __restrict__ pmax)
{
    // 16B-aligned LDS tiles: async b128 writes require 16B-aligned LDS addresses
    __shared__ alignas(16) float s_u[3][TY + 2][N];   // x-1, x, x+1 planes with y-halo
    __shared__ alignas(16) float s_im[TY][N];
    __shared__ float red[THREADS];

    const int tid = threadIdx.x;
    const int blk = blockIdx.x;
    const int b   = blk / BLOCKS_PER_SAMPLE;
    const int r   = blk % BLOCKS_PER_SAMPLE;
    const int x   = r / YT;
    const int y0  = (r % YT) * TY;

    const int xm = (x > 0)     ? x - 1 : 0;       // Neumann mirror via clamp
    const int xp = (x < N - 1) ? x + 1 : N - 1;
    const int xs0 = xm, xs1 = x, xs2 = xp;
    const size_t base = (size_t)b * NNN;

    // Stage tiles: 3 u-planes of (TY+2) rows + img tile of TY rows, float4 granularity.
    const int UPLANE4 = (TY + 2) * (N / 4);   // 576
    const int U4      = 3 * UPLANE4;          // 1728
    const int TOT4    = U4 + TY * (N / 4);    // 2240
    for (int i = tid; i < TOT4; i += THREADS) {
        if (i < U4) {
            const int p   = i / UPLANE4;                 // plane 0..2
            const int q   = i % UPLANE4;
            const int row = q >> 5;                      // 0..TY+1
            const int z4  = (q & 31) << 2;               // 0,4,...,124
            int yy = y0 - 1 + row;
            yy = (yy < 0) ? 0 : ((yy > N - 1) ? N - 1 : yy);  // y Neumann mirror
            const int xx = (p == 0) ? xs0 : ((p == 1) ? xs1 : xs2);
            const float* g = u + base + (size_t)xx * NN + (size_t)yy * N + z4;
            async_copy16(g, &s_u[p][row][z4]);
        } else {
            const int q   = i - U4;
            const int row = q >> 5;                      // 0..TY-1
            const int z4  = (q & 31) << 2;
            const float* g = img + base + (size_t)x * NN + (size_t)(y0 + row) * N + z4;
            async_copy16(g, &s_im[row][z4]);
        }
    }
    wait_async_zero();
    __syncthreads();

    const float FDE = (float)((2.0 / 127.0) * (2.0 / 127.0) * 1.0e6); // dx^2 / eps_needle
    const float dxw = (x == 0 || x == N - 1) ? 1.0f : 2.0f;

    float lsum = 0.0f;
    float lmax = 0.0f;
    #pragma unroll
    for (int i = 0; i < (TY * N) / THREADS; ++i) {      // 8 cells / thread
        const int c   = tid + i * THREADS;
        const int ly  = c >> 7;                          // 0..TY-1
        const int z   = c & (N - 1);
        const int row = ly + 1;
        const int zm  = (z > 0)     ? z - 1 : 0;
        const int zp  = (z < N - 1) ? z + 1 : N - 1;

        const float uc  = s_u[1][row][z];
        const float lap = s_u[1][row][zm] + s_u[1][row][zp]
                        + s_u[1][row - 1][z] + s_u[1][row + 1][z]
                        + s_u[0][row][z] + s_u[2][row][z]
                        - 6.0f * uc;

        const bool fz = (s_im[ly][z] != 0.0f);           // f == 1
        lsum += fz ? 0.0f : fabsf(lap);                  // |Au - b|*(1-f) collapses to this

        const int   y    = y0 + ly;
        const float diag = dxw
                         + ((y == 0 || y == N - 1) ? 1.0f : 2.0f)
                         + ((z == 0 || z == N - 1) ? 1.0f : 2.0f);
        lmax = fmaxf(lmax, diag + (fz ? FDE : 0.0f));
    }

    // deterministic block tree reductions
    red[tid] = lsum; __syncthreads();
    for (int s = THREADS / 2; s > 0; s >>= 1) {
        if (tid < s) red[tid] += red[tid + s];
        __syncthreads();
    }
    if (tid == 0) psum[blk] = red[0];
    __syncthreads();

    red[tid] = lmax; __syncthreads();
    for (int s = THREADS / 2; s > 0; s >>= 1) {
        if (tid < s) red[tid] = fmaxf(red[tid], red[tid + s]);
        __syncthreads();
    }
    if (tid == 0) pmax[blk] = red[0];
}

// ---- Finalize: per-sample S/M, global mean (deterministic, f64 accumulation) ----
__global__ __launch_bounds__(THREADS) void lossu_finalize_kernel(
        const float* __restrict__ psum, const float* __restrict__ pmax,
        float* __restrict__ out)
{
    __shared__ double rs[THREADS];
    __shared__ float  rm[THREADS];
    const int tid = threadIdx.x;
    double total = 0.0;

    for (int b = 0; b < BATCH; ++b) {
        double s = 0.0; float m = 0.0f;
        for (int i = tid; i < BLOCKS_PER_SAMPLE; i += THREADS) {
            s += (double)psum[b * BLOCKS_PER_SAMPLE + i];
            m  = fmaxf(m, pmax[b * BLOCKS_PER_SAMPLE + i]);
        }
        rs[tid] = s; rm[tid] = m; __syncthreads();
        for (int st = THREADS / 2; st > 0; st >>= 1) {
            if (tid < st) {
                rs[tid] += rs[tid + st];
                rm[tid]  = fmaxf(rm[tid], rm[tid + st]);
            }
            __syncthreads();
        }
        if (tid == 0) total += rs[0] / (double)rm[0];
        __syncthreads();
    }
    if (tid == 0) out[0] = (float)(total / (double)((size_t)BATCH * (size_t)NNN));
}

extern "C" void kernel_launch(void* const* d_in, const int* in_sizes, int n_in,
                              void* d_out, int out_size, void* d_ws, size_t ws_size,
                              hipStream_t stream) {
    // setup_inputs order: {"output": u, "gt": (unused by reference), "input": img}
    const float* u   = (const float*)d_in[0];
    const float* img = (const float*)d_in[2];
    float* psum = (float*)d_ws;            // NBLOCKS floats
    float* pmax = psum + NBLOCKS;          // NBLOCKS floats  (32 KiB total)

    lossu_stencil_kernel<<<NBLOCKS, THREADS, 0, stream>>>(u, img, psum, pmax);
    lossu_finalize_kernel<<<1, THREADS, 0, stream>>>(psum, pmax, (float*)d_out);
}